// TreeAttention_48447231099510
// MI455X (gfx1250) — compile-verified
//
#include <hip/hip_runtime.h>
#include <math.h>

typedef float v2f __attribute__((ext_vector_type(2)));
typedef float v8f __attribute__((ext_vector_type(8)));

constexpr int HEADS  = 8;
constexpr int SEQ    = 4096;
constexpr int DIM    = 128;
constexpr int TDENSE = 1024;
constexpr int KTOP   = 128;
constexpr int NW     = 8;      // waves per sparse block

#if defined(__has_builtin)
#if __has_builtin(__builtin_amdgcn_global_load_async_to_lds_b128)
#define USE_ASYNC_LDS 1
#endif
#endif

__device__ __forceinline__ v8f wmma_f32_16x16x4(v2f a, v2f b, v8f c) {
  // D = A(16x4 f32) * B(4x16 f32) + C(16x16 f32)
  return __builtin_amdgcn_wmma_f32_16x16x4_f32(false, a, false, b, (short)0, c,
                                               false, false);
}

#ifdef USE_ASYNC_LDS
typedef int v4i_vs __attribute__((vector_size(4 * sizeof(int))));
typedef __attribute__((address_space(1))) v4i_vs* gb_v4i_ptr;
typedef __attribute__((address_space(3))) v4i_vs* ld_v4i_ptr;

__device__ __forceinline__ void async_b128(const float* g, float* l) {
  __builtin_amdgcn_global_load_async_to_lds_b128(
      (gb_v4i_ptr)(void*)g, (ld_v4i_ptr)(void*)l, 0, 0);
}
__device__ __forceinline__ void wait_async0() {
#if __has_builtin(__builtin_amdgcn_s_wait_asynccnt)
  __builtin_amdgcn_s_wait_asynccnt(0);
#else
  asm volatile("s_wait_asynccnt 0x0" ::: "memory");
#endif
}
#endif

// order-preserving float<->u32 mapping (bigger float -> bigger uint)
__device__ __forceinline__ unsigned int f2mono(float f) {
  unsigned int x = __float_as_uint(f);
  return (x & 0x80000000u) ? ~x : (x | 0x80000000u);
}
__device__ __forceinline__ float mono2f(unsigned int u) {
  unsigned int x = (u & 0x80000000u) ? (u & 0x7fffffffu) : ~u;
  return __uint_as_float(x);
}

__device__ __forceinline__ float hmax16(float v) {
  v = fmaxf(v, __shfl_xor(v, 1, 16));
  v = fmaxf(v, __shfl_xor(v, 2, 16));
  v = fmaxf(v, __shfl_xor(v, 4, 16));
  v = fmaxf(v, __shfl_xor(v, 8, 16));
  return v;
}
__device__ __forceinline__ float hsum16(float v) {
  v += __shfl_xor(v, 1, 16);
  v += __shfl_xor(v, 2, 16);
  v += __shfl_xor(v, 4, 16);
  v += __shfl_xor(v, 8, 16);
  return v;
}

// ---------------------------------------------------------------------------
// Dense causal flash-attention for queries [0, TDENSE).
// One wave per 16-query tile per head; K/V tiles double-buffered through LDS
// via async-to-LDS DMA (overlapped with WMMA of the current tile).
// ---------------------------------------------------------------------------
__global__ __launch_bounds__(32) void tree_attn_dense(
    const float* __restrict__ q, const float* __restrict__ k,
    const float* __restrict__ v, float* __restrict__ out) {
  __shared__ float ldsK[2][16 * 132];  // padded row stride (132)
  __shared__ float ldsV[2][16 * 132];
  __shared__ float ldsP[16 * 20];

  const int h    = blockIdx.y;
  const int q0   = blockIdx.x * 16;
  const int lane = threadIdx.x & 31;
  const int half = lane >> 4;
  const int n16  = lane & 15;

  const float* qh = q + (size_t)h * SEQ * DIM;
  const float* kh = k + (size_t)h * SEQ * DIM;
  const float* vh = v + (size_t)h * SEQ * DIM;
  float*       oh = out + (size_t)h * SEQ * DIM;

  // Q fragments, A-matrix layout: lane m=n16, K offset half*2 + {0,1}
  v2f qa[32];
  {
    const float* qrow = qh + (size_t)(q0 + n16) * DIM;
#pragma unroll
    for (int kb = 0; kb < 32; ++kb)
      qa[kb] = *(const v2f*)(qrow + kb * 4 + half * 2);
  }

  const v8f zero8 = {0.f, 0.f, 0.f, 0.f, 0.f, 0.f, 0.f, 0.f};
  v8f acc[8];
#pragma unroll
  for (int c = 0; c < 8; ++c) acc[c] = zero8;
  float rowm[8], rowl[8];
#pragma unroll
  for (int r = 0; r < 8; ++r) { rowm[r] = -__builtin_inff(); rowl[r] = 0.f; }

  const int ktmax = q0 / 16;  // diagonal key tile

#ifdef USE_ASYNC_LDS
  // prologue: async-stage tile 0 into buffer 0 (16B per lane per row)
#pragma unroll 4
  for (int i = 0; i < 16; ++i) {
    async_b128(kh + (size_t)i * DIM + lane * 4, &ldsK[0][i * 132 + lane * 4]);
    async_b128(vh + (size_t)i * DIM + lane * 4, &ldsV[0][i * 132 + lane * 4]);
  }
  wait_async0();
#endif

  for (int kt = 0; kt <= ktmax; ++kt) {
    const int buf = kt & 1;

#ifdef USE_ASYNC_LDS
    // stage next tile into the other buffer while we compute on this one
    if (kt < ktmax) {
      const int nt = kt + 1;
#pragma unroll 4
      for (int i = 0; i < 16; ++i) {
        async_b128(kh + (size_t)(nt * 16 + i) * DIM + lane * 4,
                   &ldsK[buf ^ 1][i * 132 + lane * 4]);
        async_b128(vh + (size_t)(nt * 16 + i) * DIM + lane * 4,
                   &ldsV[buf ^ 1][i * 132 + lane * 4]);
      }
    }
#else
    // synchronous staging fallback
#pragma unroll 4
    for (int i = 0; i < 16; ++i) {
      const float4 kv4 = ((const float4*)(kh + (size_t)(kt * 16 + i) * DIM))[lane];
      const float4 vv4 = ((const float4*)(vh + (size_t)(kt * 16 + i) * DIM))[lane];
      ((float4*)(&ldsK[buf][i * 132]))[lane] = kv4;
      ((float4*)(&ldsV[buf][i * 132]))[lane] = vv4;
    }
    __syncthreads();
#endif

    // S = Q * K^T for this 16x16 tile, K-dim = 128 in steps of 4
    v8f s = zero8;
#pragma unroll
    for (int kb = 0; kb < 32; ++kb) {
      v2f b = *(const v2f*)(&ldsK[buf][n16 * 132 + kb * 4 + half * 2]);
      s = wmma_f32_16x16x4(qa[kb], b, s);
    }

    if (kt == ktmax) {  // causal mask on the diagonal tile
#pragma unroll
      for (int r = 0; r < 8; ++r)
        if (kt * 16 + n16 > q0 + r + half * 8) s[r] = -__builtin_inff();
    }

    // online softmax update (C-layout: VGPR r holds rows r / r+8)
#pragma unroll
    for (int r = 0; r < 8; ++r) {
      const float mx   = hmax16(s[r]);
      const float mnew = fmaxf(rowm[r], mx);
      const float sc   = __expf(rowm[r] - mnew);
      const float p    = __expf(s[r] - mnew);
      const float rs   = hsum16(p);
      rowm[r] = mnew;
      rowl[r] = rowl[r] * sc + rs;
#pragma unroll
      for (int c = 0; c < 8; ++c) acc[c][r] *= sc;
      ldsP[(r + half * 8) * 20 + n16] = p;  // C-layout -> LDS
    }

    // P(16x16) * V(16x128): P re-read in A-layout, V as B fragments
    v2f pa[4];
#pragma unroll
    for (int ks = 0; ks < 4; ++ks)
      pa[ks] = *(const v2f*)(&ldsP[n16 * 20 + ks * 4 + half * 2]);

#pragma unroll
    for (int c = 0; c < 8; ++c) {
#pragma unroll
      for (int ks = 0; ks < 4; ++ks) {
        const int kk = ks * 4 + half * 2;
        v2f b;
        b.x = ldsV[buf][kk * 132 + c * 16 + n16];
        b.y = ldsV[buf][(kk + 1) * 132 + c * 16 + n16];
        acc[c] = wmma_f32_16x16x4(pa[ks], b, acc[c]);
      }
    }

#ifdef USE_ASYNC_LDS
    wait_async0();  // next tile's buffer is now resident
#else
    __syncthreads();
#endif
  }

#pragma unroll
  for (int r = 0; r < 8; ++r) {
    const float inv = 1.f / rowl[r];
    float* orow = oh + (size_t)(q0 + r + half * 8) * DIM;
#pragma unroll
    for (int c = 0; c < 8; ++c) orow[c * 16 + n16] = acc[c][r] * inv;
  }
}

// ---------------------------------------------------------------------------
// Sparse top-K attention for queries [TDENSE, SEQ).
// 8 waves per 16-query tile; two 8-row passes to bound LDS at 136 KB.
// ---------------------------------------------------------------------------
__global__ __launch_bounds__(256) void tree_attn_sparse(
    const float* __restrict__ q, const float* __restrict__ k,
    const float* __restrict__ v, float* __restrict__ out) {
  __shared__ unsigned int uscore[8][SEQ];  // 128 KB, monotonic-coded scores
  __shared__ unsigned int hist[NW][256];   // 8 KB radix histograms

  const int h    = blockIdx.y;
  const int q0   = TDENSE + blockIdx.x * 16;
  const int wave = threadIdx.x >> 5;
  const int lane = threadIdx.x & 31;
  const int half = lane >> 4;
  const int n16  = lane & 15;

  const float* qh = q + (size_t)h * SEQ * DIM;
  const float* kh = k + (size_t)h * SEQ * DIM;
  const float* vh = v + (size_t)h * SEQ * DIM;
  float*       oh = out + (size_t)h * SEQ * DIM;

  const int nkt   = q0 / 16 + 1;  // key tiles covering keys 0..q0+15
  const int total = nkt * 16;

  // per-wave Q fragments (persist across both passes)
  v2f qa[32];
  {
    const float* qrow = qh + (size_t)(q0 + n16) * DIM;
#pragma unroll
    for (int kb = 0; kb < 32; ++kb)
      qa[kb] = *(const v2f*)(qrow + kb * 4 + half * 2);
  }
  const v8f zero8 = {0.f, 0.f, 0.f, 0.f, 0.f, 0.f, 0.f, 0.f};

  for (int pass = 0; pass < 2; ++pass) {
    // ---- Phase A: scores via WMMA, keep rows [pass*8, pass*8+8) ----
    for (int kt = wave; kt < nkt; kt += NW) {
      v8f s = zero8;
      const float* kbase = kh + (size_t)(kt * 16 + n16) * DIM + half * 2;
#pragma unroll
      for (int kb = 0; kb < 32; ++kb) {
        v2f b = *(const v2f*)(kbase + kb * 4);
        s = wmma_f32_16x16x4(qa[kb], b, s);
      }
      const int kj = kt * 16 + n16;
#pragma unroll
      for (int r = 0; r < 8; ++r) {
        const int row = r + half * 8;
        if ((row >> 3) == pass) {
          const unsigned int u = (kj > q0 + row) ? 0u : f2mono(s[r]);
          uscore[row & 7][kj] = u;
        }
      }
    }
    __syncthreads();

    // ---- Phase B: one row per wave ----
    {
      const int qi = q0 + pass * 8 + wave;
      const unsigned int* us = uscore[wave];

      // row max (monotonic coding preserves order)
      unsigned int um = 0u;
      for (int j = lane; j < total; j += 32) {
        const unsigned int x = us[j];
        um = (x > um) ? x : um;
      }
#pragma unroll
      for (int m = 16; m >= 1; m >>= 1) {
        const unsigned int o = __shfl_xor(um, m, 32);
        um = (o > um) ? o : um;
      }
      const float fm = mono2f(um);

      // exact radix-select of the KTOP-th largest value
      unsigned int prefix = 0u;
      int kth = KTOP;
      for (int level = 3; level >= 0; --level) {
        const int shift = level * 8;
        for (int b = lane; b < 256; b += 32) hist[wave][b] = 0u;
        const unsigned int hi_mask =
            (level == 3) ? 0u : (0xFFFFFFFFu << (shift + 8));
        for (int j = lane; j < total; j += 32) {
          const unsigned int u = us[j];
          if ((u & hi_mask) == prefix)
            atomicAdd(&hist[wave][(u >> shift) & 0xFFu], 1u);
        }
        int below = 0, bsel = 0;
        for (int b = 255; b >= 0; --b) {
          const int c = (int)hist[wave][b];
          if (below + c >= kth) { bsel = b; break; }
          below += c;
        }
        kth -= below;
        prefix |= ((unsigned int)bsel) << shift;
      }
      const unsigned int thr = prefix;

      // count strictly-greater to know how many ties to accept
      int g = 0;
      for (int j = lane; j < total; j += 32) g += (us[j] > thr) ? 1 : 0;
#pragma unroll
      for (int m = 16; m >= 1; m >>= 1) g += __shfl_xor(g, m, 32);
      const int need_eq = KTOP - g;

      // select (stable in key order), softmax weights, gather V rows
      float4 ctx = make_float4(0.f, 0.f, 0.f, 0.f);
      float lsum = 0.f;
      int eq_seen = 0;
      for (int j0 = 0; j0 < total; j0 += 32) {
        const int j = j0 + lane;
        const unsigned int u = (j < total) ? us[j] : 0u;
        const bool eq = (u == thr);
        const unsigned int meq = (unsigned int)__ballot(eq);
        const unsigned int below_mask = (1u << lane) - 1u;
        const bool take =
            (u > thr) ||
            (eq && (eq_seen + (int)__popc(meq & below_mask) < need_eq));
        eq_seen += (int)__popc(meq);
        const float w = take ? __expf(mono2f(u) - fm) : 0.f;
        unsigned int mt = (unsigned int)__ballot(take);
        while (mt) {
          const int sl = __ffs((int)mt) - 1;
          mt &= mt - 1u;
          const float ws = __shfl(w, sl, 32);
          const float4 vv =
              ((const float4*)(vh + (size_t)(j0 + sl) * DIM))[lane];
          ctx.x += ws * vv.x;
          ctx.y += ws * vv.y;
          ctx.z += ws * vv.z;
          ctx.w += ws * vv.w;
          lsum += ws;
        }
      }
      const float inv = 1.f / lsum;
      float4 o;
      o.x = ctx.x * inv; o.y = ctx.y * inv;
      o.z = ctx.z * inv; o.w = ctx.w * inv;
      ((float4*)(oh + (size_t)qi * DIM))[lane] = o;
    }
    __syncthreads();  // before next pass overwrites uscore
  }
}

extern "C" void kernel_launch(void* const* d_in, const int* in_sizes, int n_in,
                              void* d_out, int out_size, void* d_ws,
                              size_t ws_size, hipStream_t stream) {
  (void)in_sizes; (void)n_in; (void)out_size; (void)d_ws; (void)ws_size;
  const float* q = (const float*)d_in[0];
  const float* k = (const float*)d_in[1];
  const float* v = (const float*)d_in[2];
  float* out = (float*)d_out;

  dim3 gd(TDENSE / 16, HEADS);
  tree_attn_dense<<<gd, 32, 0, stream>>>(q, k, v, out);

  dim3 gs((SEQ - TDENSE) / 16, HEADS);
  tree_attn_sparse<<<gs, 256, 0, stream>>>(q, k, v, out);
}